// RelGraphConv_64493228917358
// MI455X (gfx1250) — compile-verified
//
#include <hip/hip_runtime.h>

#define N_NODES   20000
#define N_EDGES   320000
#define IN_DIM    256
#define OUT_DIM   256
#define NUM_RELS  32
#define NUM_BASES 8

typedef float v2f __attribute__((ext_vector_type(2)));
typedef float v4f __attribute__((ext_vector_type(4)));
typedef float v8f __attribute__((ext_vector_type(8)));

// ---------------------------------------------------------------------------
// Stage 0: zero the output accumulator (we scatter-add into d_out directly).
// ---------------------------------------------------------------------------
__global__ void zero_f32(float* __restrict__ p, int n) {
  int i = blockIdx.x * blockDim.x + threadIdx.x;
  if (i < n) p[i] = 0.0f;
}

// ---------------------------------------------------------------------------
// Stage 1 (fast path): yb[n][b][o] = sum_i h[n][i] * bases[b][i][o]
// 8 independent GEMMs, M=20000 (1250 16-row tiles), N=256, K=256.
// One wave computes a 16x64 slab: 4 accumulators share one A fragment per
// K-step of 4 (V_WMMA_F32_16X16X4_F32).
//
// f32 fragment layouts (ISA 7.12.2):
//   A 16x4 : v0 = {lanes0-15: K=0, lanes16-31: K=2}, v1 = {K=1, K=3}; M = lane&15
//   B 4x16 : v0 = {lanes0-15: K=0 row, lanes16-31: K=2 row}, v1 = {K=1, K=3}; N = lane&15
//   C 16x16: VGPR v -> rows M=v (lanes0-15) and M=v+8 (lanes16-31), N = lane&15
// ---------------------------------------------------------------------------
__global__ void __launch_bounds__(128)
yb_gemm_wmma(const float* __restrict__ h,      // [N_NODES][IN_DIM]
             const float* __restrict__ bases,  // [NUM_BASES][IN_DIM][OUT_DIM]
             float* __restrict__ yb)           // [N_NODES][NUM_BASES][OUT_DIM]
{
  const int lane = threadIdx.x & 31;
  const int wave = threadIdx.x >> 5;   // 0..3 -> which 64-column slab
  const int l15  = lane & 15;
  const int half = lane >> 4;          // 0/1: which K-pair this lane carries
  const int mtile = blockIdx.x;        // 0..1249
  const int b     = blockIdx.y;        // 0..7

  const float* __restrict__ hrow = h + (size_t)(mtile * 16 + l15) * IN_DIM;
  const float* __restrict__ bb   = bases + (size_t)b * IN_DIM * OUT_DIM;
  const int colbase = wave * 64;

  v8f acc0 = {}, acc1 = {}, acc2 = {}, acc3 = {};

  #pragma unroll 4
  for (int k0 = 0; k0 < IN_DIM; k0 += 4) {
    const int kk = k0 + half * 2;
    v2f a;
    a.x = hrow[kk];
    a.y = hrow[kk + 1];

    const float* __restrict__ br0 = bb + (size_t)kk * OUT_DIM + colbase + l15;
    const float* __restrict__ br1 = br0 + OUT_DIM;
    v2f b0; b0.x = br0[0];  b0.y = br1[0];
    v2f b1; b1.x = br0[16]; b1.y = br1[16];
    v2f b2; b2.x = br0[32]; b2.y = br1[32];
    v2f b3; b3.x = br0[48]; b3.y = br1[48];

    acc0 = __builtin_amdgcn_wmma_f32_16x16x4_f32(false, a, false, b0, (short)0, acc0, false, false);
    acc1 = __builtin_amdgcn_wmma_f32_16x16x4_f32(false, a, false, b1, (short)0, acc1, false, false);
    acc2 = __builtin_amdgcn_wmma_f32_16x16x4_f32(false, a, false, b2, (short)0, acc2, false, false);
    acc3 = __builtin_amdgcn_wmma_f32_16x16x4_f32(false, a, false, b3, (short)0, acc3, false, false);
  }

  const size_t nodeStride = (size_t)NUM_BASES * OUT_DIM;  // 2048 floats / node
  #pragma unroll
  for (int v = 0; v < 8; ++v) {
    const int r = mtile * 16 + half * 8 + v;
    const size_t base = (size_t)r * nodeStride + (size_t)b * OUT_DIM + colbase + l15;
    yb[base +  0] = acc0[v];
    yb[base + 16] = acc1[v];
    yb[base + 32] = acc2[v];
    yb[base + 48] = acc3[v];
  }
}

// ---------------------------------------------------------------------------
// Stage 2 (fast path): per-edge rank-8 combine + scatter-add.
// One wave per edge; lane owns outputs [lane*8, lane*8+8). yb row (8 KB) is
// L2-resident (yb = 164 MB < 192 MB L2).
// ---------------------------------------------------------------------------
__global__ void __launch_bounds__(256)
edge_combine_scatter(const float* __restrict__ yb,
                     const int*   __restrict__ src,
                     const int*   __restrict__ dst,
                     const int*   __restrict__ rel,
                     const float* __restrict__ norm,
                     const float* __restrict__ w_comp,  // [NUM_RELS][NUM_BASES]
                     float* __restrict__ out)
{
  const int e = blockIdx.x * 8 + (threadIdx.x >> 5);
  if (e >= N_EDGES) return;
  const int lane = threadIdx.x & 31;

  const int   s  = src[e];
  const int   d  = dst[e];
  const int   r  = rel[e];
  const float nm = norm[e];

  const float* __restrict__ ys = yb + (size_t)s * (NUM_BASES * OUT_DIM) + lane * 8;
  const float* __restrict__ wc = w_comp + r * NUM_BASES;

  float acc[8] = {0.f, 0.f, 0.f, 0.f, 0.f, 0.f, 0.f, 0.f};
  #pragma unroll
  for (int b = 0; b < NUM_BASES; ++b) {
    const float c = wc[b];
    const v4f* p = (const v4f*)(ys + b * OUT_DIM);
    const v4f u0 = p[0];
    const v4f u1 = p[1];
    acc[0] += c * u0.x; acc[1] += c * u0.y; acc[2] += c * u0.z; acc[3] += c * u0.w;
    acc[4] += c * u1.x; acc[5] += c * u1.y; acc[6] += c * u1.z; acc[7] += c * u1.w;
  }

  float* op = out + (size_t)d * OUT_DIM + lane * 8;
  #pragma unroll
  for (int j = 0; j < 8; ++j) atomicAdd(op + j, acc[j] * nm);
}

// ---------------------------------------------------------------------------
// Stage 3: in-place ReLU after all scatter-adds.
// ---------------------------------------------------------------------------
__global__ void relu_f32(float* __restrict__ p, int n) {
  int i = blockIdx.x * blockDim.x + threadIdx.x;
  if (i < n) {
    float v = p[i];
    p[i] = v > 0.0f ? v : 0.0f;
  }
}

// ---------------------------------------------------------------------------
// Fallback A (ws >= 8 MB): materialize weight[r] = sum_b w_comp[r,b]*bases[b],
// then per-edge matvec msg = h[src] @ W[rel] with h row staged in LDS.
// ---------------------------------------------------------------------------
__global__ void combine_weight(const float* __restrict__ bases,
                               const float* __restrict__ w_comp,
                               float* __restrict__ W) {
  int idx = blockIdx.x * blockDim.x + threadIdx.x;
  if (idx >= NUM_RELS * IN_DIM * OUT_DIM) return;
  const int r  = idx >> 16;          // / (256*256)
  const int io = idx & 0xFFFF;
  float acc = 0.f;
  #pragma unroll
  for (int b = 0; b < NUM_BASES; ++b)
    acc += w_comp[r * NUM_BASES + b] * bases[(size_t)b * IN_DIM * OUT_DIM + io];
  W[idx] = acc;
}

__global__ void __launch_bounds__(256)
edge_matvec(const float* __restrict__ h,
            const int* __restrict__ src, const int* __restrict__ dst,
            const int* __restrict__ rel, const float* __restrict__ norm,
            const float* __restrict__ W, float* __restrict__ out) {
  __shared__ float sh[IN_DIM];
  const int e = blockIdx.x;
  const int t = threadIdx.x;
  sh[t] = h[(size_t)src[e] * IN_DIM + t];
  __syncthreads();
  const float* __restrict__ wr = W + (size_t)rel[e] * IN_DIM * OUT_DIM;
  float acc = 0.f;
  #pragma unroll 8
  for (int i = 0; i < IN_DIM; ++i) acc += sh[i] * wr[(size_t)i * OUT_DIM + t];
  atomicAdd(out + (size_t)dst[e] * OUT_DIM + t, acc * norm[e]);
}

// ---------------------------------------------------------------------------
// Fallback B (no usable scratch): per-edge compute straight from bases.
// ---------------------------------------------------------------------------
__global__ void __launch_bounds__(256)
edge_direct(const float* __restrict__ h,
            const int* __restrict__ src, const int* __restrict__ dst,
            const int* __restrict__ rel, const float* __restrict__ norm,
            const float* __restrict__ bases, const float* __restrict__ w_comp,
            float* __restrict__ out) {
  __shared__ float sh[IN_DIM];
  const int e = blockIdx.x;
  const int t = threadIdx.x;
  sh[t] = h[(size_t)src[e] * IN_DIM + t];
  __syncthreads();
  const int r = rel[e];
  float acc = 0.f;
  for (int b = 0; b < NUM_BASES; ++b) {
    const float c = w_comp[r * NUM_BASES + b];
    const float* __restrict__ bb = bases + (size_t)b * IN_DIM * OUT_DIM + t;
    float ab = 0.f;
    #pragma unroll 8
    for (int i = 0; i < IN_DIM; ++i) ab += sh[i] * bb[(size_t)i * OUT_DIM];
    acc += c * ab;
  }
  atomicAdd(out + (size_t)dst[e] * OUT_DIM + t, acc * norm[e]);
}

// ---------------------------------------------------------------------------
// Host launcher
// ---------------------------------------------------------------------------
extern "C" void kernel_launch(void* const* d_in, const int* in_sizes, int n_in,
                              void* d_out, int out_size, void* d_ws, size_t ws_size,
                              hipStream_t stream) {
  const float* h      = (const float*)d_in[0];
  const int*   src    = (const int*)  d_in[1];
  const int*   dst    = (const int*)  d_in[2];
  const int*   rel    = (const int*)  d_in[3];
  const float* norm   = (const float*)d_in[4];
  const float* bases  = (const float*)d_in[5];
  const float* w_comp = (const float*)d_in[6];
  float* out = (float*)d_out;

  const int nOut = N_NODES * OUT_DIM;
  zero_f32<<<(nOut + 255) / 256, 256, 0, stream>>>(out, nOut);

  const size_t YB_BYTES = (size_t)N_NODES * NUM_BASES * OUT_DIM * sizeof(float);
  const size_t W_BYTES  = (size_t)NUM_RELS * IN_DIM * OUT_DIM * sizeof(float);

  if (ws_size >= YB_BYTES) {
    float* yb = (float*)d_ws;
    dim3 grid(N_NODES / 16, NUM_BASES);
    yb_gemm_wmma<<<grid, 128, 0, stream>>>(h, bases, yb);
    edge_combine_scatter<<<N_EDGES / 8, 256, 0, stream>>>(yb, src, dst, rel, norm, w_comp, out);
  } else if (ws_size >= W_BYTES) {
    float* W = (float*)d_ws;
    combine_weight<<<(NUM_RELS * IN_DIM * OUT_DIM + 255) / 256, 256, 0, stream>>>(bases, w_comp, W);
    edge_matvec<<<N_EDGES, 256, 0, stream>>>(h, src, dst, rel, norm, W, out);
  } else {
    edge_direct<<<N_EDGES, 256, 0, stream>>>(h, src, dst, rel, norm, bases, w_comp, out);
  }

  relu_f32<<<(nOut + 255) / 256, 256, 0, stream>>>(out, nOut);
}